// GraphEncoder_21165598835040
// MI455X (gfx1250) — compile-verified
//
#include <hip/hip_runtime.h>
#include <hip/hip_bf16.h>

typedef float v2f __attribute__((ext_vector_type(2)));
typedef float v8f __attribute__((ext_vector_type(8)));

#define D_IN 512
#define DH   256
#define NTYPES 8
#define NREL 6
#define KT 64

// ---------------- small helper kernels ----------------

__global__ void degree_kernel(const int* __restrict__ dst, float* __restrict__ deg, int E) {
    int i = blockIdx.x * blockDim.x + threadIdx.x;
    if (i < E) atomicAdd(&deg[dst[i]], 1.0f);
}

__global__ void invdeg_kernel(const float* __restrict__ deg, float* __restrict__ inv, int N) {
    int i = blockIdx.x * blockDim.x + threadIdx.x;
    if (i < N) inv[i] = 1.0f / fmaxf(deg[i], 1.0f);
}

__global__ void hist_kernel(const int* __restrict__ nt, int* __restrict__ counts, int N) {
    int i = blockIdx.x * blockDim.x + threadIdx.x;
    if (i < N) atomicAdd(&counts[nt[i]], 1);
}

// single-thread: padded (16-aligned) bucket offsets; offs[8] = padded total
__global__ void offsets_kernel(const int* __restrict__ counts, int* __restrict__ offs) {
    if (blockIdx.x == 0 && threadIdx.x == 0) {
        int off = 0;
        for (int t = 0; t < NTYPES; ++t) {
            offs[t] = off;
            off += ((counts[t] + 15) >> 4) << 4;
        }
        offs[NTYPES] = off;
    }
}

__global__ void build_perm_kernel(const int* __restrict__ nt, const int* __restrict__ offs,
                                  int* __restrict__ fill, int* __restrict__ perm, int N) {
    int i = blockIdx.x * blockDim.x + threadIdx.x;
    if (i >= N) return;
    int t = nt[i];
    int pos = offs[t] + atomicAdd(&fill[t], 1);
    perm[pos] = i;
}

// ---------------- adapt GEMM: h = tanh(W_t x + b_t), type-bucketed tiles ----------------
// grid.x = node tiles (16 nodes, homogeneous type via perm), grid.y = DH/64, block = 128 (4 waves)
__global__ __launch_bounds__(128)
void adapt_gemm_kernel(const float* __restrict__ X, const float* __restrict__ Wt,
                       const float* __restrict__ bias, const int* __restrict__ node_type,
                       const int* __restrict__ perm, const int* __restrict__ offs,
                       float* __restrict__ H) {
    __shared__ float Xs[16][KT + 1];
    __shared__ float Ws[KT][64 + 1];
    __shared__ int   perm_s[16];
    __shared__ int   type_s;

    int tile = blockIdx.x;
    if (tile * 16 >= offs[NTYPES]) return;          // uniform early exit past padded total

    int tid = threadIdx.x;
    if (tid < 16) perm_s[tid] = perm[tile * 16 + tid];
    __syncthreads();
    if (tid == 0) type_s = node_type[perm_s[0]];    // first row of a tile is always a real node
    __syncthreads();

    int t = type_s;
    const float* W = Wt + (size_t)t * DH * D_IN;    // row-major [DH][D_IN] -> need B[k][o] = W[o][k]
    int wave = tid >> 5, lane = tid & 31;
    int m = lane & 15;
    int koff = (lane < 16) ? 0 : 2;
    int obase = blockIdx.y * 64;
    int o = obase + wave * 16 + m;

    v8f c = {};
    for (int k0 = 0; k0 < D_IN; k0 += KT) {
        // stage 16x64 A tile (gathered node rows)
        {
            int r = tid >> 3, c0 = (tid & 7) * 8;
            int node = perm_s[r];
            const float* src = X + (size_t)(node < 0 ? 0 : node) * D_IN + k0 + c0;
            #pragma unroll
            for (int j = 0; j < 8; ++j) Xs[r][c0 + j] = (node >= 0) ? src[j] : 0.0f;
        }
        // stage 64x64 W tile transposed: Ws[k][o_local] = W[obase+o_local][k0+k]
        {
            int ol = tid >> 1, kp = (tid & 1) * 32;
            const float* src = W + (size_t)(obase + ol) * D_IN + k0 + kp;
            #pragma unroll
            for (int j = 0; j < 32; ++j) Ws[kp + j][ol] = src[j];
        }
        __syncthreads();
        // batch all K-tile operands into registers (one clustered LDS clause, one wait),
        // then run 16 back-to-back accumulation-chained WMMAs
        float aX[16], aY[16], bX[16], bY[16];
        #pragma unroll
        for (int q = 0; q < 16; ++q) {
            int kk = q * 4 + koff;
            aX[q] = Xs[m][kk];
            aY[q] = Xs[m][kk + 1];
            bX[q] = Ws[kk][wave * 16 + m];
            bY[q] = Ws[kk + 1][wave * 16 + m];
        }
        #pragma unroll
        for (int q = 0; q < 16; ++q) {
            v2f a = {aX[q], aY[q]};
            v2f b = {bX[q], bY[q]};
            c = __builtin_amdgcn_wmma_f32_16x16x4_f32(false, a, false, b, (short)0, c, false, false);
        }
        __syncthreads();
    }

    float bv = bias[t * DH + o];
    #pragma unroll
    for (int v = 0; v < 8; ++v) {
        int row = (lane < 16) ? v : v + 8;
        int node = perm_s[row];
        if (node >= 0) H[(size_t)node * DH + o] = tanhf(c[v] + bv);
    }
}

// ---------------- k-major GEMM: Out[n][e] = sum_d A[n][d] * W[d][e] (W k-major, L2 resident) ----
// grid.x = ceil(N/16), grid.y = DH/64, block = 128
__global__ __launch_bounds__(128)
void gemm_kmajor_kernel(const float* __restrict__ A, const float* __restrict__ W,
                        float* __restrict__ Out, int N) {
    __shared__ float Xs[16][KT + 1];
    int n0 = blockIdx.x * 16;
    int tid = threadIdx.x, wave = tid >> 5, lane = tid & 31;
    int m = lane & 15;
    int koff = (lane < 16) ? 0 : 2;
    int o = blockIdx.y * 64 + wave * 16 + m;

    v8f c = {};
    for (int k0 = 0; k0 < DH; k0 += KT) {
        int r = tid >> 3, c0 = (tid & 7) * 8;
        int node = n0 + r;
        const float* src = A + (size_t)(node < N ? node : 0) * DH + k0 + c0;
        #pragma unroll
        for (int j = 0; j < 8; ++j) Xs[r][c0 + j] = (node < N) ? src[j] : 0.0f;
        __syncthreads();
        // batch: 32 clustered global B loads (L2-resident weights) + LDS A loads,
        // single wait, then 16 chained WMMAs
        float aX[16], aY[16], bX[16], bY[16];
        #pragma unroll
        for (int q = 0; q < 16; ++q) {
            int kk = q * 4 + koff;
            int k = k0 + kk;
            aX[q] = Xs[m][kk];
            aY[q] = Xs[m][kk + 1];
            bX[q] = W[(size_t)k * DH + o];        // coalesced: lanes vary o
            bY[q] = W[(size_t)(k + 1) * DH + o];
        }
        #pragma unroll
        for (int q = 0; q < 16; ++q) {
            v2f a = {aX[q], aY[q]};
            v2f b = {bX[q], bY[q]};
            c = __builtin_amdgcn_wmma_f32_16x16x4_f32(false, a, false, b, (short)0, c, false, false);
        }
        __syncthreads();
    }
    #pragma unroll
    for (int v = 0; v < 8; ++v) {
        int row = (lane < 16) ? v : v + 8;
        int node = n0 + row;
        if (node < N) Out[(size_t)node * DH + o] = c[v];
    }
}

// ---------------- edge scatter for one relation: agg[dst] += hr[src] ----------------
// one wave32 per edge; 8 f32 atomics per lane (L2-resident agg)
__global__ __launch_bounds__(256)
void scatter_rel_kernel(const float* __restrict__ hr, const int* __restrict__ src,
                        const int* __restrict__ dst, const int* __restrict__ etype,
                        int rel, float* __restrict__ agg, int E) {
    int wave = blockIdx.x * (blockDim.x >> 5) + (threadIdx.x >> 5);
    int lane = threadIdx.x & 31;
    if (wave >= E) return;
    if (etype[wave] != rel) return;
    int s = src[wave], d = dst[wave];
    const float4* hp = (const float4*)(hr + (size_t)s * DH);
    float* ap = agg + (size_t)d * DH;
    #pragma unroll
    for (int j = 0; j < 2; ++j) {
        int idx = lane + j * 32;
        float4 v = hp[idx];
        int base = idx * 4;
        atomicAdd(ap + base + 0, v.x);
        atomicAdd(ap + base + 1, v.y);
        atomicAdd(ap + base + 2, v.z);
        atomicAdd(ap + base + 3, v.w);
    }
}

// ---------------- combine: h = relu(agg * inv_deg + selfout + b_self) ----------------
__global__ void combine_kernel(const float* __restrict__ agg, const float* __restrict__ selfout,
                               const float* __restrict__ invdeg, const float* __restrict__ bself,
                               float* __restrict__ hout, int N) {
    int i = blockIdx.x * blockDim.x + threadIdx.x;
    int total = N * (DH / 4);
    if (i >= total) return;
    int n  = i / (DH / 4);
    int o4 = i % (DH / 4);
    float4 a = ((const float4*)agg)[i];
    float4 s = ((const float4*)selfout)[i];
    float4 b = ((const float4*)bself)[o4];
    float id = invdeg[n];
    float4 r;
    r.x = fmaxf(a.x * id + s.x + b.x, 0.0f);
    r.y = fmaxf(a.y * id + s.y + b.y, 0.0f);
    r.z = fmaxf(a.z * id + s.z + b.z, 0.0f);
    r.w = fmaxf(a.w * id + s.w + b.w, 0.0f);
    ((float4*)hout)[i] = r;
}

// ---------------- host launch ----------------

extern "C" void kernel_launch(void* const* d_in, const int* in_sizes, int n_in,
                              void* d_out, int out_size, void* d_ws, size_t ws_size,
                              hipStream_t stream) {
    const float* node_feature = (const float*)d_in[0];
    const int*   node_type    = (const int*)d_in[1];
    const int*   edge_index   = (const int*)d_in[2];
    const int*   edge_type    = (const int*)d_in[3];
    // d_in[4] = node_position (unused by reference)
    const float* adapt_W      = (const float*)d_in[5];
    const float* adapt_b      = (const float*)d_in[6];
    const float* W_rel        = (const float*)d_in[7];
    const float* W_self       = (const float*)d_in[8];
    const float* b_self       = (const float*)d_in[9];

    const int N = in_sizes[0] / D_IN;
    const int E = in_sizes[3];
    const int* src = edge_index;       // row 0
    const int* dst = edge_index + E;   // row 1

    // workspace carve-out (256B aligned slabs)
    char* ws = (char*)d_ws;
    size_t off = 0;
    auto carve = [&](size_t bytes) -> char* {
        char* p = ws + off;
        off = (off + bytes + 255) & ~(size_t)255;
        return p;
    };
    float* h0      = (float*)carve((size_t)N * DH * 4);
    float* h1      = (float*)carve((size_t)N * DH * 4);
    float* hr      = (float*)carve((size_t)N * DH * 4);
    float* selfout = (float*)carve((size_t)N * DH * 4);
    float* agg     = (float*)carve((size_t)N * DH * 4);
    float* deg     = (float*)carve((size_t)N * 4);
    float* invdeg  = (float*)carve((size_t)N * 4);
    int*   perm    = (int*)carve((size_t)(N + NTYPES * 16) * 4);
    int*   counts  = (int*)carve(NTYPES * 4);
    int*   fill    = (int*)carve(NTYPES * 4);
    int*   offs    = (int*)carve((NTYPES + 1) * 4);
    (void)ws_size;

    hipMemsetAsync(deg,    0,    (size_t)N * 4, stream);
    hipMemsetAsync(agg,    0,    (size_t)N * DH * 4, stream);
    hipMemsetAsync(counts, 0,    NTYPES * 4, stream);
    hipMemsetAsync(fill,   0,    NTYPES * 4, stream);
    hipMemsetAsync(perm,   0xFF, (size_t)(N + NTYPES * 16) * 4, stream);

    // degree / inverse degree
    degree_kernel<<<(E + 255) / 256, 256, 0, stream>>>(dst, deg, E);
    invdeg_kernel<<<(N + 255) / 256, 256, 0, stream>>>(deg, invdeg, N);

    // counting sort of nodes by type (padded to 16-node tiles)
    hist_kernel<<<(N + 255) / 256, 256, 0, stream>>>(node_type, counts, N);
    offsets_kernel<<<1, 32, 0, stream>>>(counts, offs);
    build_perm_kernel<<<(N + 255) / 256, 256, 0, stream>>>(node_type, offs, fill, perm, N);

    // type-adapted projection -> h0
    int tiles_max = (N + NTYPES * 16 + 15) / 16;    // >= padded total / 16; device exits early
    adapt_gemm_kernel<<<dim3(tiles_max, DH / 64), 128, 0, stream>>>(
        node_feature, adapt_W, adapt_b, node_type, perm, offs, h0);

    // two relational layers
    const int ntile = (N + 15) / 16;
    float* hin = h0;
    for (int l = 0; l < 2; ++l) {
        gemm_kmajor_kernel<<<dim3(ntile, DH / 64), 128, 0, stream>>>(
            hin, W_self + (size_t)l * DH * DH, selfout, N);
        for (int r = 0; r < NREL; ++r) {
            gemm_kmajor_kernel<<<dim3(ntile, DH / 64), 128, 0, stream>>>(
                hin, W_rel + ((size_t)l * NREL + r) * DH * DH, hr, N);
            scatter_rel_kernel<<<(E + 7) / 8, 256, 0, stream>>>(
                hr, src, dst, edge_type, r, agg, E);
        }
        float* hout = (l == 0) ? h1 : (float*)d_out;
        combine_kernel<<<((size_t)N * (DH / 4) + 255) / 256, 256, 0, stream>>>(
            agg, selfout, invdeg, b_self + (size_t)l * DH, hout, N);
        if (l == 0) {
            hipMemsetAsync(agg, 0, (size_t)N * DH * 4, stream);
            hin = h1;
        }
    }
}